// EditDistStatModel_4930622455858
// MI455X (gfx1250) — compile-verified
//
#include <hip/hip_runtime.h>
#include <math.h>

#define NCLS 676
#define BIG_NEG (-1e30f)

typedef float v2f __attribute__((ext_vector_type(2)));
typedef float v8f __attribute__((ext_vector_type(8)));

// Numerically safe logaddexp (inputs may be ~ -1e30 but never NaN).
__device__ __forceinline__ float lae(float a, float b) {
    float m = fmaxf(a, b);
    float d = fabsf(a - b);
    return m + log1pf(expf(-d));
}

struct AddOp { __device__ float operator()(float a, float b) const { return a + b; } };
struct LaeOp { __device__ float operator()(float a, float b) const { return lae(a, b); } };

// Inclusive block-wide scan over 256 threads (8 wave32 waves):
// intra-wave shuffle scan (depth 5) + 8-partial combine in LDS.
template <typename Op>
__device__ __forceinline__ float block_scan(float x, float* lds8, int tid, Op op) {
    int lane = tid & 31, wv = tid >> 5;
    #pragma unroll
    for (int d = 1; d < 32; d <<= 1) {
        float n = __shfl_up(x, d, 32);
        if (lane >= d) x = op(x, n);
    }
    if (lane == 31) lds8[wv] = x;
    __syncthreads();
    if (wv == 0 && lane < 8) {
        float v = lds8[lane];
        #pragma unroll
        for (int d = 1; d < 8; d <<= 1) {
            float n = __shfl_up(v, d, 32);
            if (lane >= d) v = op(v, n);
        }
        lds8[lane] = v;
    }
    __syncthreads();
    if (wv > 0) x = op(x, lds8[wv - 1]);
    __syncthreads();   // protect lds8 for next call
    return x;
}

// Block 0: forward alpha.  Block 1: backward beta (reference's source-indexed weights).
__global__ __launch_bounds__(256) void k_alphabeta(const int* __restrict__ ar,
                                                   const int* __restrict__ en,
                                                   const float* __restrict__ w,
                                                   float* __restrict__ alpha,
                                                   float* __restrict__ beta) {
    __shared__ float w_lds[NCLS];
    __shared__ int   ar_lds[256];
    __shared__ float row[256];
    __shared__ float lds8[8];
    int tid = threadIdx.x;
    for (int k = tid; k < NCLS; k += 256) w_lds[k] = w[k];
    ar_lds[tid] = ar[tid];
    __syncthreads();

    if (blockIdx.x == 0) {
        int j = tid;
        int enj = en[j];
        float wins = w_lds[26 + enj];
        // s[j] = sum_{k=1..j} w_ins[k]  (same for every row)
        float s = block_scan((j == 0) ? 0.f : wins, lds8, tid, AddOp());
        // row 0: chain of c0 = [0, BIG_NEG...]
        float c = (j == 0) ? 0.f : BIG_NEG;
        float a = s + block_scan(c - s, lds8, tid, LaeOp());
        row[j] = a; alpha[j] = a;
        __syncthreads();
        for (int i = 1; i < 256; ++i) {
            int ai = ar_lds[i];
            float wd = w_lds[1 + ai];
            float wsub = w_lds[51 + 25 * ai + enj];
            float pj   = row[j];
            float pjm1 = (j > 0) ? row[j - 1] : 0.f;
            float c2 = (j == 0) ? (wd + pj) : lae(wd + pj, wsub + pjm1);
            float a2 = s + block_scan(c2 - s, lds8, tid, LaeOp());
            row[j] = a2; alpha[i * 256 + j] = a2;
            __syncthreads();
        }
    } else {
        int t = tid, j = 255 - t;          // reversed thread<->column mapping
        int enj = en[j];
        float wins = w_lds[26 + enj];      // == m_rev[t]
        float s = block_scan((t == 0) ? 0.f : wins, lds8, tid, AddOp());
        float c = (t == 0) ? 0.f : BIG_NEG;  // cT reversed
        float b = s + block_scan(c - s, lds8, tid, LaeOp());
        row[j] = b; beta[255 * 256 + j] = b;
        __syncthreads();
        for (int i = 254; i >= 0; --i) {
            int ai = ar_lds[i];
            float wd = w_lds[1 + ai];
            float wsub = w_lds[51 + 25 * ai + enj];
            float nj   = row[j];
            float njp1 = (j < 255) ? row[j + 1] : 0.f;
            float c2 = (j == 255) ? (wd + nj) : lae(wd + nj, wsub + njp1);
            float b2 = s + block_scan(c2 - s, lds8, tid, LaeOp());
            row[j] = b2; beta[i * 256 + j] = b2;
            __syncthreads();
        }
    }
}

// D[i] = lse_j(alpha[i-1][j]+beta[i][j]); I[j] = lse_i(alpha[i][j-1]+beta[i][j]);
// G = max over i,j>=1 of alpha[i-1][j-1]+beta[i][j]   (shift for the sub GEMM)
__global__ __launch_bounds__(256) void k_rowcol(const float* __restrict__ alpha,
                                                const float* __restrict__ beta,
                                                float* __restrict__ D,
                                                float* __restrict__ I,
                                                float* __restrict__ Gout) {
    __shared__ float red[256];
    int t = threadIdx.x;
    if (t >= 1) {
        float mx = -INFINITY;
        for (int j = 0; j < 256; ++j) mx = fmaxf(mx, alpha[(t - 1) * 256 + j] + beta[t * 256 + j]);
        float sm = 0.f;
        for (int j = 0; j < 256; ++j) sm += expf(alpha[(t - 1) * 256 + j] + beta[t * 256 + j] - mx);
        D[t] = mx + logf(sm);
    } else D[0] = -INFINITY;
    if (t >= 1) {
        float mx = -INFINITY;
        for (int i = 0; i < 256; ++i) mx = fmaxf(mx, alpha[i * 256 + t - 1] + beta[i * 256 + t]);
        float sm = 0.f;
        for (int i = 0; i < 256; ++i) sm += expf(alpha[i * 256 + t - 1] + beta[i * 256 + t] - mx);
        I[t] = mx + logf(sm);
    } else I[0] = -INFINITY;
    float g = -INFINITY;
    if (t >= 1)
        for (int j = 1; j < 256; ++j)
            g = fmaxf(g, alpha[(t - 1) * 256 + (j - 1)] + beta[t * 256 + j]);
    red[t] = g; __syncthreads();
    for (int k = 128; k > 0; k >>= 1) {
        if (t < k) red[t] = fmaxf(red[t], red[t + k]);
        __syncthreads();
    }
    if (t == 0) Gout[0] = red[0];
}

// P[i][j] = exp(alpha[i-1][j-1]+beta[i][j]-G) (zero-padded row/col 0),
// plus one-hot matrices Aoh (32 x 256, rows=ar classes) and Eoh (256 x 32).
__global__ __launch_bounds__(256) void k_buildP(const float* __restrict__ alpha,
                                                const float* __restrict__ beta,
                                                const int* __restrict__ ar,
                                                const int* __restrict__ en,
                                                const float* __restrict__ Gp,
                                                float* __restrict__ P,
                                                float* __restrict__ Aoh,
                                                float* __restrict__ Eoh) {
    int i = blockIdx.x, j = threadIdx.x;
    float G = Gp[0];
    float v = 0.f;
    if (i >= 1 && j >= 1)
        v = expf(alpha[(i - 1) * 256 + (j - 1)] + beta[i * 256 + j] - G);
    P[i * 256 + j] = v;
    if (i == 0) {
        int aj = ar[j], ej = en[j];
        for (int a = 0; a < 32; ++a) Aoh[a * 256 + j] = (j >= 1 && a == aj) ? 1.f : 0.f;
        for (int e = 0; e < 32; ++e) Eoh[j * 32 + e]  = (j >= 1 && e == ej) ? 1.f : 0.f;
    }
}

// Sub-class posteriors as two f32 WMMA GEMMs:
//   R(32x256) = Aoh(32x256) @ P(256x256);  S(32x32) = R @ Eoh(256x32)
// out[51+25a+e] = w + G + log(S[a][e])
__global__ __launch_bounds__(256) void k_subgemm(const float* __restrict__ P,
                                                 const float* __restrict__ Aoh,
                                                 const float* __restrict__ Eoh,
                                                 const float* __restrict__ w,
                                                 const float* __restrict__ Gp,
                                                 float* __restrict__ out) {
    __shared__ float Rl[32 * 256];
    int tid  = threadIdx.x;
    int lane = tid & 31, wv = tid >> 5;
    int hi = lane >> 4;        // lane half (ISA frag layout)
    int ln = lane & 15;

    // GEMM1: 2x16 = 32 output tiles of 16x16, 4 per wave; K=256 in steps of 4.
    for (int q = 0; q < 4; ++q) {
        int tl = wv + 8 * q;
        int mi = tl >> 4, ni = tl & 15;
        v8f acc = {};
        for (int kk = 0; kk < 64; ++kk) {
            int k0 = kk * 4;
            int ka = k0 + hi * 2;
            v2f af, bf;
            // A 16x4 f32 frag: lanes 0-15 -> K=k0,k0+1 ; lanes 16-31 -> K=k0+2,k0+3
            af.x = Aoh[(mi * 16 + ln) * 256 + ka];
            af.y = Aoh[(mi * 16 + ln) * 256 + ka + 1];
            // B 4x16 f32 frag: VGPR v holds rows v (lanes 0-15) and v+2 (lanes 16-31)
            bf.x = P[ka * 256 + ni * 16 + ln];
            bf.y = P[(ka + 1) * 256 + ni * 16 + ln];
            acc = __builtin_amdgcn_wmma_f32_16x16x4_f32(false, af, false, bf,
                                                        (short)0, acc, false, false);
        }
        #pragma unroll
        for (int v = 0; v < 8; ++v) {   // C/D layout: VGPR v -> rows v / v+8
            int m = mi * 16 + v + hi * 8;
            Rl[m * 256 + ni * 16 + ln] = acc[v];
        }
    }
    __syncthreads();

    // GEMM2: 2x2 tiles, waves 0..3
    if (wv < 4) {
        int mi = wv >> 1, ni = wv & 1;
        v8f acc = {};
        for (int kk = 0; kk < 64; ++kk) {
            int k0 = kk * 4;
            int ka = k0 + hi * 2;
            v2f af, bf;
            af.x = Rl[(mi * 16 + ln) * 256 + ka];
            af.y = Rl[(mi * 16 + ln) * 256 + ka + 1];
            bf.x = Eoh[ka * 32 + ni * 16 + ln];
            bf.y = Eoh[(ka + 1) * 32 + ni * 16 + ln];
            acc = __builtin_amdgcn_wmma_f32_16x16x4_f32(false, af, false, bf,
                                                        (short)0, acc, false, false);
        }
        float G = Gp[0];
        #pragma unroll
        for (int v = 0; v < 8; ++v) {
            int a = mi * 16 + v + hi * 8;
            int e = ni * 16 + ln;
            if (a < 25 && e < 25) {
                int c = 51 + 25 * a + e;
                float sv = acc[v];
                out[c] = (sv > 0.f) ? (w[c] + G + logf(sv)) : -INFINITY;
            }
        }
    }
}

// Classes 0..50 (null / del / ins) — per-class stable lse over D / I.
__global__ __launch_bounds__(64) void k_delins(const int* __restrict__ ar,
                                               const int* __restrict__ en,
                                               const float* __restrict__ w,
                                               const float* __restrict__ D,
                                               const float* __restrict__ I,
                                               float* __restrict__ out) {
    int c = threadIdx.x;
    if (c >= 51) return;
    if (c == 0) { out[0] = -INFINITY; return; }
    float mx = -INFINITY;
    if (c <= 25) {
        int sym = c - 1;
        for (int i = 1; i < 256; ++i) if (ar[i] == sym) mx = fmaxf(mx, D[i]);
        if (mx == -INFINITY) { out[c] = -INFINITY; return; }
        float s = 0.f;
        for (int i = 1; i < 256; ++i) if (ar[i] == sym) s += expf(D[i] - mx);
        out[c] = w[c] + mx + logf(s);
    } else {
        int sym = c - 26;
        for (int j = 1; j < 256; ++j) if (en[j] == sym) mx = fmaxf(mx, I[j]);
        if (mx == -INFINITY) { out[c] = -INFINITY; return; }
        float s = 0.f;
        for (int j = 1; j < 256; ++j) if (en[j] == sym) s += expf(I[j] - mx);
        out[c] = w[c] + mx + logf(s);
    }
}

extern "C" void kernel_launch(void* const* d_in, const int* in_sizes, int n_in,
                              void* d_out, int out_size, void* d_ws, size_t ws_size,
                              hipStream_t stream) {
    const int*   ar = (const int*)d_in[0];
    const int*   en = (const int*)d_in[1];
    const float* w  = (const float*)d_in[2];
    float* out = (float*)d_out;
    float* ws  = (float*)d_ws;

    float* alpha = ws;                 // 65536
    float* beta  = ws + 65536;         // 65536
    float* P     = ws + 131072;        // 65536
    float* Aoh   = ws + 196608;        // 8192
    float* Eoh   = ws + 204800;        // 8192
    float* D     = ws + 212992;        // 256
    float* I     = ws + 213248;        // 256
    float* G     = ws + 213504;        // 1

    k_alphabeta<<<2, 256, 0, stream>>>(ar, en, w, alpha, beta);
    k_rowcol  <<<1, 256, 0, stream>>>(alpha, beta, D, I, G);
    k_buildP  <<<256, 256, 0, stream>>>(alpha, beta, ar, en, G, P, Aoh, Eoh);
    k_subgemm <<<1, 256, 0, stream>>>(P, Aoh, Eoh, w, G, out);
    k_delins  <<<1, 64, 0, stream>>>(ar, en, w, D, I, out);
}